// DescriptionEmbedding_36988258353169
// MI455X (gfx1250) — compile-verified
//
#include <hip/hip_runtime.h>
#include <hip/hip_bf16.h>

// ---------------------------------------------------------------------------
// MI455X (gfx1250) fused attention-embedding pipeline.
// All matrix ops via V_WMMA_F32_16X16X4_F32 (native f32 WMMA, wave32).
// Workspace (floats):
//   pre_f   [512  x 64]                        (padded to 16-row tiles)
//   pre_wb  [2512 x 64]   = full@W2 + bw       (padded)
//   S       [512  x 2500] masked exp(score)    (rows 500..511 zeroed)
//   ssum    [512]         row sums (0 -> 1)
//   ctx     [512  x 64]   context rows
// Total ~6.03 MB -> L2 resident (192 MB).
// ---------------------------------------------------------------------------

typedef __attribute__((ext_vector_type(2))) float v2f;
typedef __attribute__((ext_vector_type(8))) float v8f;

#define WMMA_F32X4(a, b, c) \
  __builtin_amdgcn_wmma_f32_16x16x4_f32(false, (a), false, (b), (short)0, (c), false, false)

constexpr int Fd   = 500;
constexpr int Hd   = 2000;
constexpr int FULL = 2500;   // F + H
constexpr int Dd   = 64;
constexpr int Ad   = 64;
constexpr int Bd   = 4096;

constexpr int WTILES = (FULL + 15) / 16;  // 157
constexpr int FTILES = (Fd + 15) / 16;    // 32
constexpr int MTILES = Bd / 16;           // 256
constexpr int CTILES = (Fd + 15) / 16;    // 32

__device__ __forceinline__ float tanh_fast(float x) {
  // tanh(x) = 1 - 2 / (exp(2x) + 1); v_exp_f32 is exp2 -> scale by 2*log2(e).
  // Saturates to +/-1 at large |x| (inf -> rcp -> 0), no NaN from inf/inf.
  float e = __builtin_amdgcn_exp2f(x * 2.8853900817779268f);
  return 1.0f - 2.0f * __builtin_amdgcn_rcpf(e + 1.0f);
}

__device__ __forceinline__ v2f load_v2(const float* p) {
  return *reinterpret_cast<const v2f*>(p);   // 8B-aligned -> global_load_b64
}

// ---------------------------------------------------------------------------
// Kernel 1: pre_f = feat @ W1 ; pre_wb = concat(feat,hid) @ W2 + bw
// One 16-row M-tile per block; 4 waves, one 16-col N-tile per wave.
// blocks [0, WTILES)             -> pre_wb tiles (with bias)
// blocks [WTILES, WTILES+FTILES) -> pre_f tiles
// Row guard is branchless: load from a clamped (always valid) row, then
// zero the fragment with v_cndmask. No exec-mask branches in the WMMA loop.
// ---------------------------------------------------------------------------
__global__ __launch_bounds__(128) void proj_kernel(
    const float* __restrict__ feat, const float* __restrict__ hid,
    const float* __restrict__ Ww,   const float* __restrict__ bw,
    float* __restrict__ pre_f, float* __restrict__ pre_wb)
{
  const int lane = threadIdx.x & 31;
  const int l15  = lane & 15;
  const int half = lane >> 4;           // K sub-offset (ISA A-frag layout)
  const int n0   = (threadIdx.x >> 5) * 16;

  const bool isW   = (blockIdx.x < WTILES);
  const int  mt    = isW ? (int)blockIdx.x : (int)blockIdx.x - WTILES;
  const int  row   = mt * 16 + l15;     // A-matrix row fed by this lane
  const int  limit = isW ? FULL : Fd;
  const bool valid = row < limit;
  const int  rc    = valid ? row : (limit - 1);   // clamped, always in-range

  // Per-lane A row pointer (hoisted; lowers to one v_cndmask_b64 pair).
  const float* arow = (isW && rc >= Fd) ? (hid + (rc - Fd) * Dd)
                                        : (feat + rc * Dd);
  const float* Wcol = Ww + (isW ? Dd * Ad : 0) + n0 + l15;  // B column base
  const float  zmul = valid ? 1.0f : 0.0f;                  // branchless zero

  v8f acc = {};
  #pragma unroll
  for (int k = 0; k < Dd / 4; ++k) {
    const int kb = k * 4 + 2 * half;
    v2f a = load_v2(arow + kb);
    a.x *= zmul; a.y *= zmul;
    v2f b;
    b.x = Wcol[kb * Ad];
    b.y = Wcol[(kb + 1) * Ad];
    acc = WMMA_F32X4(a, b, acc);
  }

  const int   col  = n0 + l15;
  const float bias = isW ? bw[col] : 0.0f;
  #pragma unroll
  for (int i = 0; i < 8; ++i) {
    const int   r = mt * 16 + i + 8 * half;   // C/D frag: VGPR i -> M = i + 8*half
    const float v = acc[i] + bias;
    if (isW) { if (r < FULL) pre_wb[r * Ad + col] = v; }
    else     { if (r < Fd)   pre_f [r * Ad + col] = v; }
  }
}

// ---------------------------------------------------------------------------
// Kernel 2: S[f,j] = mask ? exp( Wu . tanh(pre_f[f] + pre_wb[j]) ) : 0
//           ssum[f] = sum_j S[f,j]  (0 -> 1)
// One block per f (blocks >= F zero the S padding rows for kernel 3).
// This is the FLOP-heavy part: 80M v_exp_f32 trans ops, fused, h never hits
// memory (vs 320 MB if materialized).
// ---------------------------------------------------------------------------
__global__ __launch_bounds__(256) void score_kernel(
    const float* __restrict__ pre_f, const float* __restrict__ pre_wb,
    const float* __restrict__ Wu,    const unsigned char* __restrict__ mask,
    float* __restrict__ S, float* __restrict__ ssum)
{
  const int f   = blockIdx.x;
  const int tid = threadIdx.x;

  if (f >= Fd) {                    // zero padding rows 500..511 of S
    for (int j = tid; j < FULL; j += 256) S[(size_t)f * FULL + j] = 0.0f;
    if (tid == 0) ssum[f] = 1.0f;
    return;
  }

  __shared__ float pf[Ad];
  __shared__ float wu[Ad];
  __shared__ float red[256];
  if (tid < Ad) { pf[tid] = pre_f[f * Ad + tid]; wu[tid] = Wu[tid]; }
  __syncthreads();

  constexpr float LOG2E = 1.4426950408889634f;
  float local = 0.0f;
  for (int j = tid; j < FULL; j += 256) {
    const float* pw = pre_wb + (size_t)j * Ad;
    __builtin_prefetch(pw + 256 * Ad, 0, 0);    // this thread's next j-row
    float sc = 0.0f;
    #pragma unroll
    for (int q = 0; q < Ad / 4; ++q) {
      const float4 pv = reinterpret_cast<const float4*>(pw)[q];
      const int a = q * 4;
      sc += tanh_fast(pf[a + 0] + pv.x) * wu[a + 0];
      sc += tanh_fast(pf[a + 1] + pv.y) * wu[a + 1];
      sc += tanh_fast(pf[a + 2] + pv.z) * wu[a + 2];
      sc += tanh_fast(pf[a + 3] + pv.w) * wu[a + 3];
    }
    const float s = mask[(size_t)f * FULL + j]
                      ? __builtin_amdgcn_exp2f(sc * LOG2E) : 0.0f;
    S[(size_t)f * FULL + j] = s;
    local += s;
  }

  red[tid] = local;
  __syncthreads();
  #pragma unroll
  for (int off = 128; off > 0; off >>= 1) {
    if (tid < off) red[tid] += red[tid + off];
    __syncthreads();
  }
  if (tid == 0) { const float s = red[0]; ssum[f] = (s == 0.0f) ? 1.0f : s; }
}

// ---------------------------------------------------------------------------
// Kernel 3: ctx = (S / ssum) @ full     (500x2500)@(2500x64), K = 625 * 4
// One M-tile per block, one N-tile per wave. S rows padded+zeroed -> no K
// guard. Unroll x5 (625 = 5*125) so loads clause and multiple load->WMMA
// pairs stay in flight.
// ---------------------------------------------------------------------------
__global__ __launch_bounds__(128) void context_kernel(
    const float* __restrict__ S, const float* __restrict__ feat,
    const float* __restrict__ hid, const float* __restrict__ ssum,
    float* __restrict__ ctx)
{
  const int lane = threadIdx.x & 31;
  const int l15  = lane & 15;
  const int half = lane >> 4;
  const int n0   = (threadIdx.x >> 5) * 16;
  const int mt   = blockIdx.x;

  const float* srow = S + (size_t)(mt * 16 + l15) * FULL;

  v8f acc = {};
  #pragma unroll 5
  for (int k = 0; k < FULL / 4; ++k) {
    const int kb = k * 4 + 2 * half;
    __builtin_prefetch(srow + k * 4 + 128, 0, 0);   // stream S row ahead
    v2f a = load_v2(srow + kb);
    const float* r0 = (kb     < Fd) ? (feat + kb * Dd)       : (hid + (kb - Fd) * Dd);
    const float* r1 = (kb + 1 < Fd) ? (feat + (kb + 1) * Dd) : (hid + (kb + 1 - Fd) * Dd);
    v2f b; b.x = r0[n0 + l15]; b.y = r1[n0 + l15];
    acc = WMMA_F32X4(a, b, acc);
  }

  #pragma unroll
  for (int i = 0; i < 8; ++i) {
    const int r = mt * 16 + i + 8 * half;
    if (r < Fd) ctx[r * Dd + n0 + l15] = acc[i] / ssum[r];
  }
}

// ---------------------------------------------------------------------------
// Kernel 4: out = values @ ctx          (4096x500)@(500x64), K = 125 * 4
// M = 4096 = 256 tiles exactly -> no guards anywhere. Unroll x5 (125 = 5*25).
// ---------------------------------------------------------------------------
__global__ __launch_bounds__(128) void out_kernel(
    const float* __restrict__ values, const float* __restrict__ ctx,
    float* __restrict__ out)
{
  const int lane = threadIdx.x & 31;
  const int l15  = lane & 15;
  const int half = lane >> 4;
  const int n0   = (threadIdx.x >> 5) * 16;
  const int mt   = blockIdx.x;

  const float* vrow = values + (size_t)(mt * 16 + l15) * Fd;
  const float* ccol = ctx + n0 + l15;

  v8f acc = {};
  #pragma unroll 5
  for (int k = 0; k < Fd / 4; ++k) {
    const int kb = k * 4 + 2 * half;
    v2f a = load_v2(vrow + kb);
    v2f b; b.x = ccol[kb * Dd]; b.y = ccol[(kb + 1) * Dd];
    acc = WMMA_F32X4(a, b, acc);
  }

  #pragma unroll
  for (int i = 0; i < 8; ++i) {
    const int r = mt * 16 + i + 8 * half;
    out[r * Dd + n0 + l15] = acc[i];
  }
}

// ---------------------------------------------------------------------------
extern "C" void kernel_launch(void* const* d_in, const int* in_sizes, int n_in,
                              void* d_out, int out_size, void* d_ws, size_t ws_size,
                              hipStream_t stream) {
  const float*         values = (const float*)d_in[0];          // (4096, 500)
  const float*         feat   = (const float*)d_in[1];          // (500, 64)
  const float*         hid    = (const float*)d_in[2];          // (2000, 64)
  const float*         Ww     = (const float*)d_in[3];          // (128, 64)
  const float*         bw     = (const float*)d_in[4];          // (64,)
  const float*         Wu     = (const float*)d_in[5];          // (64, 1)
  const unsigned char* mask   = (const unsigned char*)d_in[6];  // (500, 2500) bool
  float*               out    = (float*)d_out;                  // (4096, 64)

  float* ws     = (float*)d_ws;
  float* pre_f  = ws;                     // 512*64
  float* pre_wb = pre_f + 512 * Ad;       // 2512*64
  float* S      = pre_wb + 2512 * Ad;     // 512*2500
  float* ssum   = S + 512 * FULL;         // 512
  float* ctx    = ssum + 512;             // 512*64

  proj_kernel   <<<WTILES + FTILES, 128, 0, stream>>>(feat, hid, Ww, bw, pre_f, pre_wb);
  score_kernel  <<<512,             256, 0, stream>>>(pre_f, pre_wb, Wu, mask, S, ssum);
  context_kernel<<<CTILES,          128, 0, stream>>>(S, feat, hid, ssum, ctx);
  out_kernel    <<<MTILES,          128, 0, stream>>>(values, ctx, out);
}